// SMPL_Layer_35570919145863
// MI455X (gfx1250) — compile-verified
//
#include <hip/hip_runtime.h>

// ---------------------------------------------------------------------------
// SMPL forward for MI455X (gfx1250, wave32).
// Big GEMMs (posedirs @ pose_map, weights @ G_rel) run on the CDNA5 matrix
// pipeline via V_WMMA_F32_16X16X4_F32 (fp32 in/out, matches reference dtype).
// Each wave owns 4 output tiles sharing one operand's fetches (4x FLOP per
// load), and the K tail is handled branch-free with clamped loads + selects.
// ---------------------------------------------------------------------------

typedef float v2f __attribute__((ext_vector_type(2)));
typedef float v8f __attribute__((ext_vector_type(8)));

constexpr int Bn = 1024;
constexpr int Vn = 6890;
constexpr int NJ = 24;
constexpr int M3 = Vn * 3;      // 20670  (m = v*3 + c)
constexpr int KP = 207;         // pose_map length

// d_out layout: verts, jtr, v_posed, naked, G  (flat, in return order)
constexpr size_t OFF_VERTS  = 0;
constexpr size_t OFF_JTR    = (size_t)Bn * Vn * 3;                 // 21166080
constexpr size_t OFF_VPOSED = OFF_JTR + (size_t)Bn * NJ * 3;       // 21239808
constexpr size_t OFF_NAKED  = OFF_VPOSED + (size_t)Bn * Vn * 3;    // 42405888
constexpr size_t OFF_G      = OFF_NAKED + (size_t)Bn * Vn * 3;     // 63571968

// workspace (floats): pose_map | j | G_rel   (~2.7 MB)
constexpr size_t WS_PM   = 0;
constexpr size_t WS_J    = (size_t)Bn * KP;                        // 211968
constexpr size_t WS_GREL = WS_J + (size_t)Bn * NJ * 3;             // +73728

__constant__ int PAR[NJ] = {-1, 0, 0, 0, 1, 2, 3, 4, 5, 6, 7, 8,
                            9, 9, 9, 12, 13, 14, 16, 17, 18, 19, 20, 21};

// ---------------------------------------------------------------------------
// k1: v_shaped[b, m] = v_template[m] + sum_s shapedirs[m, s] * betas[b, s]
//     (written into the v_posed output region as scratch; k5 overwrites it)
// ---------------------------------------------------------------------------
__global__ void k1_shape(const float* __restrict__ v_template,
                         const float* __restrict__ shapedirs,
                         const float* __restrict__ betas,
                         float* __restrict__ vsh) {
    size_t idx = (size_t)blockIdx.x * blockDim.x + threadIdx.x;
    if (idx >= (size_t)Bn * M3) return;
    int b = (int)(idx / M3);
    int r = (int)(idx % M3);
    const float* sd = shapedirs + (size_t)r * 10;
    const float* bt = betas + (size_t)b * 10;
    float acc = v_template[r];
#pragma unroll
    for (int s = 0; s < 10; ++s) acc += sd[s] * bt[s];
    vsh[idx] = acc;
}

// ---------------------------------------------------------------------------
// k2: j[b, jj, c] = sum_v J_regressor[jj, v] * v_shaped[b, v, c]
//     one block per (b, jj); 256-thread strided reduction + LDS tree.
// ---------------------------------------------------------------------------
__global__ void k2_jreg(const float* __restrict__ Jreg,
                        const float* __restrict__ vsh,
                        float* __restrict__ jws) {
    int b  = blockIdx.x / NJ;   // consecutive blocks share b -> L2 reuse
    int jj = blockIdx.x % NJ;
    int tid = threadIdx.x;
    __shared__ float red[3][256];
    float a0 = 0.f, a1 = 0.f, a2 = 0.f;
    const float* jr = Jreg + (size_t)jj * Vn;
    const float* vb = vsh + (size_t)b * M3;
    for (int v = tid; v < Vn; v += 256) {
        float w = jr[v];
        const float* p = vb + (size_t)v * 3;
        a0 += w * p[0]; a1 += w * p[1]; a2 += w * p[2];
    }
    red[0][tid] = a0; red[1][tid] = a1; red[2][tid] = a2;
    __syncthreads();
    for (int s = 128; s > 0; s >>= 1) {
        if (tid < s) {
            red[0][tid] += red[0][tid + s];
            red[1][tid] += red[1][tid + s];
            red[2][tid] += red[2][tid + s];
        }
        __syncthreads();
    }
    if (tid == 0) {
        float* o = jws + (size_t)b * NJ * 3 + (size_t)jj * 3;
        o[0] = red[0][0]; o[1] = red[1][0]; o[2] = red[2][0];
    }
}

// ---------------------------------------------------------------------------
// k3: one wave per batch. Rodrigues per lane(=joint), pose_map, serial
//     kinematic chain, G (out), jtr = G[:, :3, 3] + trans (out), G_rel (ws).
// ---------------------------------------------------------------------------
__global__ __launch_bounds__(32) void k3_kin(const float* __restrict__ pose,
                                             const float* __restrict__ trans,
                                             const float* __restrict__ jws,
                                             float* __restrict__ pose_map,
                                             float* __restrict__ g_rel,
                                             float* __restrict__ out) {
    int b = blockIdx.x, lane = threadIdx.x;
    __shared__ float R[NJ][9];
    __shared__ float jp[NJ][3];
    __shared__ float G[NJ][12];   // rows 0..8: 3x3 rotation, 9..11: translation
    if (lane < NJ) {
        float rx = pose[(size_t)b * 72 + lane * 3 + 0];
        float ry = pose[(size_t)b * 72 + lane * 3 + 1];
        float rz = pose[(size_t)b * 72 + lane * 3 + 2];
        float rr  = rx * rx + ry * ry + rz * rz + 1e-16f;
        float ang = sqrtf(rr);
        float inv = 1.0f / ang;
        float x = rx * inv, y = ry * inv, z = rz * inv;
        float c = cosf(ang), s = sinf(ang), t = 1.0f - c;
        float r0 = c + t * x * x,   r1 = t * x * y - s * z, r2 = t * x * z + s * y;
        float r3 = t * x * y + s * z, r4 = c + t * y * y,   r5 = t * y * z - s * x;
        float r6 = t * x * z - s * y, r7 = t * y * z + s * x, r8 = c + t * z * z;
        R[lane][0] = r0; R[lane][1] = r1; R[lane][2] = r2;
        R[lane][3] = r3; R[lane][4] = r4; R[lane][5] = r5;
        R[lane][6] = r6; R[lane][7] = r7; R[lane][8] = r8;
        jp[lane][0] = jws[(size_t)b * 72 + lane * 3 + 0];
        jp[lane][1] = jws[(size_t)b * 72 + lane * 3 + 1];
        jp[lane][2] = jws[(size_t)b * 72 + lane * 3 + 2];
        if (lane > 0) {
            float* pm = pose_map + (size_t)b * KP + (size_t)(lane - 1) * 9;
            pm[0] = r0 - 1.f; pm[1] = r1;       pm[2] = r2;
            pm[3] = r3;       pm[4] = r4 - 1.f; pm[5] = r5;
            pm[6] = r6;       pm[7] = r7;       pm[8] = r8 - 1.f;
        }
    }
    __syncthreads();
    if (lane == 0) {
#pragma unroll
        for (int k = 0; k < 9; ++k) G[0][k] = R[0][k];
        G[0][9] = jp[0][0]; G[0][10] = jp[0][1]; G[0][11] = jp[0][2];
        for (int i = 1; i < NJ; ++i) {
            int p = PAR[i];
            float t0 = jp[i][0] - jp[p][0];
            float t1 = jp[i][1] - jp[p][1];
            float t2 = jp[i][2] - jp[p][2];
            float tmp[12];
#pragma unroll
            for (int r = 0; r < 3; ++r) {
                float g0 = G[p][r * 3 + 0], g1 = G[p][r * 3 + 1], g2 = G[p][r * 3 + 2];
#pragma unroll
                for (int cc = 0; cc < 3; ++cc)
                    tmp[r * 3 + cc] = g0 * R[i][cc] + g1 * R[i][3 + cc] + g2 * R[i][6 + cc];
                tmp[9 + r] = g0 * t0 + g1 * t1 + g2 * t2 + G[p][9 + r];
            }
#pragma unroll
            for (int k = 0; k < 12; ++k) G[i][k] = tmp[k];
        }
    }
    __syncthreads();
    if (lane < NJ) {
        float* go = out + OFF_G + (size_t)b * NJ * 16 + (size_t)lane * 16;
#pragma unroll
        for (int r = 0; r < 3; ++r) {
            go[r * 4 + 0] = G[lane][r * 3 + 0];
            go[r * 4 + 1] = G[lane][r * 3 + 1];
            go[r * 4 + 2] = G[lane][r * 3 + 2];
            go[r * 4 + 3] = G[lane][9 + r];
        }
        go[12] = 0.f; go[13] = 0.f; go[14] = 0.f; go[15] = 1.f;
#pragma unroll
        for (int p = 0; p < 3; ++p)
            out[OFF_JTR + (size_t)b * 72 + lane * 3 + p] = G[lane][9 + p] + trans[(size_t)b * 3 + p];
        float* gr = g_rel + (size_t)b * NJ * 16 + (size_t)lane * 16;
#pragma unroll
        for (int r = 0; r < 3; ++r) {
            float tmp = G[lane][r * 3 + 0] * jp[lane][0] +
                        G[lane][r * 3 + 1] * jp[lane][1] +
                        G[lane][r * 3 + 2] * jp[lane][2];
            gr[r * 4 + 0] = G[lane][r * 3 + 0];
            gr[r * 4 + 1] = G[lane][r * 3 + 1];
            gr[r * 4 + 2] = G[lane][r * 3 + 2];
            gr[r * 4 + 3] = G[lane][9 + r] - tmp;
        }
        gr[12] = 0.f; gr[13] = 0.f; gr[14] = 0.f; gr[15] = 1.f;
    }
}

// ---------------------------------------------------------------------------
// k4: naked[b, m] = v_shaped[b, m] + sum_p pose_map[b, p] * posedirs[m, p]
//     WMMA f32 16x16x4.  One wave = 16 batches x 64 m (4 tiles sharing the
//     A fetch).  51 branch-free K-steps + 1 select-padded tail step.
// ---------------------------------------------------------------------------
constexpr int MT4 = 4;          // m-tiles per wave in k4

__global__ __launch_bounds__(32) void k4_posegemm(const float* __restrict__ posedirs,
                                                  const float* __restrict__ pm,
                                                  const float* __restrict__ vsh,
                                                  float* __restrict__ naked) {
    int m_base = blockIdx.x * (16 * MT4);
    int b0 = blockIdx.y * 16;
    int lane = threadIdx.x;
    int hf = lane >> 4;        // K half-selector (per ISA A/B layout)
    int ln = lane & 15;        // M row (A) / N col (B)

    int mm[MT4], mmc[MT4];
    const float* pdcol[MT4];
    v8f acc[MT4];
#pragma unroll
    for (int t = 0; t < MT4; ++t) {
        mm[t]  = m_base + t * 16 + ln;
        mmc[t] = mm[t] < M3 ? mm[t] : M3 - 1;
        pdcol[t] = posedirs + (size_t)mmc[t] * KP;
        __builtin_prefetch(pdcol[t], 0, 1);                 // global_prefetch_b8
#pragma unroll
        for (int r = 0; r < 8; ++r)
            acc[t][r] = vsh[(size_t)(b0 + r + 8 * hf) * M3 + mmc[t]];
    }

    const float* pmrow = pm + (size_t)(b0 + ln) * KP;       // A row (batch)

    // main K loop: k = 0..203, no guards anywhere
#pragma unroll 2
    for (int k0 = 0; k0 < 204; k0 += 4) {
        int pa = k0 + 2 * hf;                               // pa+1 <= 203
        v2f a;
        a.x = pmrow[pa];
        a.y = pmrow[pa + 1];
#pragma unroll
        for (int t = 0; t < MT4; ++t) {
            v2f bb;
            bb.x = pdcol[t][pa];
            bb.y = pdcol[t][pa + 1];
            acc[t] = __builtin_amdgcn_wmma_f32_16x16x4_f32(
                false, a, false, bb, (short)0, acc[t], false, false);
        }
    }
    // tail: k = 204..206 valid, k = 207 zero-padded via cndmask (no branch)
    {
        int pa  = 204 + 2 * hf;                             // 204 or 206
        int pb  = pa + 1;                                   // 205 or 207
        int pbc = pb < KP ? pb : KP - 1;                    // clamp: safe load
        bool ok = pb < KP;
        v2f a;
        a.x = pmrow[pa];
        a.y = ok ? pmrow[pbc] : 0.f;
#pragma unroll
        for (int t = 0; t < MT4; ++t) {
            v2f bb;
            bb.x = pdcol[t][pa];
            bb.y = ok ? pdcol[t][pbc] : 0.f;
            acc[t] = __builtin_amdgcn_wmma_f32_16x16x4_f32(
                false, a, false, bb, (short)0, acc[t], false, false);
        }
    }

#pragma unroll
    for (int t = 0; t < MT4; ++t) {
        if (mm[t] < M3) {
#pragma unroll
            for (int r = 0; r < 8; ++r)
                naked[(size_t)(b0 + r + 8 * hf) * M3 + mm[t]] = acc[t][r];
        }
    }
}

// ---------------------------------------------------------------------------
// k5: skinning.  T[v, pq] = sum_j weights[v, j] * G_rel[b, j, pq]  (WMMA,
//     K=24 in 6 steps; 4 vertex-tiles per wave share the G_rel B fetch),
//     then verts = T(3x4) @ (posed,1) + trans.  Idle half-wave writes the
//     v_posed = naked output copy.
// ---------------------------------------------------------------------------
constexpr int VT4 = 4;          // vertex tiles per wave in k5

__global__ __launch_bounds__(32) void k5_skin(const float* __restrict__ weights,
                                              const float* __restrict__ g_rel,
                                              const float* __restrict__ trans,
                                              float* __restrict__ out) {
    int v_base = blockIdx.x * (16 * VT4);
    int b  = blockIdx.y;
    int lane = threadIdx.x;
    int hf = lane >> 4;
    int ln = lane & 15;

    const float* gb = g_rel + (size_t)b * NJ * 16;
    const float* naked = out + OFF_NAKED;

    int vc[VT4];
    v8f acc[VT4];
#pragma unroll
    for (int t = 0; t < VT4; ++t) {
        int vv = v_base + t * 16 + ln;
        vc[t] = vv < Vn ? vv : Vn - 1;
        acc[t] = {};
    }

#pragma unroll
    for (int k0 = 0; k0 < NJ; k0 += 4) {
        int ka = k0 + 2 * hf;                               // ka <= 22
        v2f bb;
        bb.x = gb[(size_t)ka * 16 + ln];
        bb.y = gb[(size_t)(ka + 1) * 16 + ln];
#pragma unroll
        for (int t = 0; t < VT4; ++t) {
            v2f a;
            a.x = weights[(size_t)vc[t] * NJ + ka];
            a.y = weights[(size_t)vc[t] * NJ + ka + 1];
            acc[t] = __builtin_amdgcn_wmma_f32_16x16x4_f32(
                false, a, false, bb, (short)0, acc[t], false, false);
        }
    }

    __shared__ float T[16 * VT4][16];   // [tile*16 + v_local][pq]
#pragma unroll
    for (int t = 0; t < VT4; ++t)
#pragma unroll
        for (int r = 0; r < 8; ++r)
            T[t * 16 + r + 8 * hf][ln] = acc[t][r];
    __syncthreads();

#pragma unroll
    for (int t = 0; t < VT4; ++t) {
        int vv = v_base + t * 16 + ln;
        if (vv < Vn) {
            const float* nb = naked + (size_t)b * M3 + (size_t)vv * 3;
            if (hf == 0) {
                float x = nb[0], y = nb[1], z = nb[2];
                const float (&Tr)[16] = T[t * 16 + ln];
#pragma unroll
                for (int p = 0; p < 3; ++p) {
                    float vp = Tr[p * 4 + 0] * x + Tr[p * 4 + 1] * y +
                               Tr[p * 4 + 2] * z + Tr[p * 4 + 3] +
                               trans[(size_t)b * 3 + p];
                    out[OFF_VERTS + (size_t)b * M3 + (size_t)vv * 3 + p] = vp;
                }
            } else {
                float* vp = out + OFF_VPOSED + (size_t)b * M3 + (size_t)vv * 3;
                vp[0] = nb[0]; vp[1] = nb[1]; vp[2] = nb[2];
            }
        }
    }
}

// ---------------------------------------------------------------------------
extern "C" void kernel_launch(void* const* d_in, const int* in_sizes, int n_in,
                              void* d_out, int out_size, void* d_ws, size_t ws_size,
                              hipStream_t stream) {
    const float* pose       = (const float*)d_in[0];
    const float* betas      = (const float*)d_in[1];
    const float* trans      = (const float*)d_in[2];
    const float* v_template = (const float*)d_in[3];
    const float* shapedirs  = (const float*)d_in[4];
    const float* posedirs   = (const float*)d_in[5];
    const float* J_regressor= (const float*)d_in[6];
    const float* weights    = (const float*)d_in[7];

    float* out = (float*)d_out;
    float* ws  = (float*)d_ws;
    float* pose_map = ws + WS_PM;
    float* jws      = ws + WS_J;
    float* g_rel    = ws + WS_GREL;

    // k1: v_shaped -> v_posed region (scratch until k5 overwrites it)
    {
        int n = (int)(((size_t)Bn * M3 + 255) / 256);
        k1_shape<<<n, 256, 0, stream>>>(v_template, shapedirs, betas, out + OFF_VPOSED);
    }
    // k2: joint regression
    k2_jreg<<<Bn * NJ, 256, 0, stream>>>(J_regressor, out + OFF_VPOSED, jws);
    // k3: rodrigues + chain -> G, jtr, G_rel, pose_map
    k3_kin<<<Bn, 32, 0, stream>>>(pose, trans, jws, pose_map, g_rel, out);
    // k4: WMMA pose-corrective GEMM -> naked (4 m-tiles per wave)
    k4_posegemm<<<dim3((M3 + 16 * MT4 - 1) / (16 * MT4), Bn / 16), 32, 0, stream>>>(
        posedirs, pose_map, out + OFF_VPOSED, out + OFF_NAKED);
    // k5: WMMA skinning -> verts (+trans) and v_posed copy (4 v-tiles per wave)
    k5_skin<<<dim3((Vn + 16 * VT4 - 1) / (16 * VT4), Bn), 32, 0, stream>>>(
        weights, g_rel, trans, out);
}